// SpanLabeler_48576080118505
// MI455X (gfx1250) — compile-verified
//
#include <hip/hip_runtime.h>

// Problem constants (from reference): B=2, L=256, D=512, H=256, O=16
#define LDIM 256
#define DDIM 512
#define HDIM 256
#define ODIM 16
#define MDIM 512   // B*L rows
#define BDIM 2
#define LEAKY 0.1f
#define EPSV 1e-9f

typedef __attribute__((ext_vector_type(2))) float v2f;
typedef __attribute__((ext_vector_type(8))) float v8f;

// ---------------------------------------------------------------------------
// Kernel 1: h = LeakyReLU(X @ W^T + b) via V_WMMA_F32_16X16X4_F32
//   X: [MDIM, DDIM] row-major, W: [HDIM, DDIM] row-major, Hout: [MDIM, HDIM]
//   One wave computes a 16(m) x 64(n) strip; K-loop steps by 4.
// ---------------------------------------------------------------------------
__global__ __launch_bounds__(32) void mlp_wmma_kernel(
    const float* __restrict__ X, const float* __restrict__ W,
    const float* __restrict__ bias, float* __restrict__ Hout)
{
    const int lane  = threadIdx.x & 31;
    const int mBase = blockIdx.x * 16;
    const int nBase = blockIdx.y * 64;
    const int mRow  = mBase + (lane & 15);
    const int kOff  = (lane >> 4) * 2;          // lanes 16-31 hold K=2,3

    v8f acc0 = {}, acc1 = {}, acc2 = {}, acc3 = {};

    const float* xrow = X + mRow * DDIM + kOff;
    const int n0 = nBase + (lane & 15);
    const float* w0 = W + (n0 +  0) * DDIM + kOff;
    const float* w1 = W + (n0 + 16) * DDIM + kOff;
    const float* w2 = W + (n0 + 32) * DDIM + kOff;
    const float* w3 = W + (n0 + 48) * DDIM + kOff;

    #pragma unroll 4
    for (int k = 0; k < DDIM; k += 4) {
        v2f a  = *(const v2f*)(xrow + k);
        v2f b0 = *(const v2f*)(w0 + k);
        v2f b1 = *(const v2f*)(w1 + k);
        v2f b2 = *(const v2f*)(w2 + k);
        v2f b3 = *(const v2f*)(w3 + k);
        acc0 = __builtin_amdgcn_wmma_f32_16x16x4_f32(false, a, false, b0, 0, acc0, false, false);
        acc1 = __builtin_amdgcn_wmma_f32_16x16x4_f32(false, a, false, b1, 0, acc1, false, false);
        acc2 = __builtin_amdgcn_wmma_f32_16x16x4_f32(false, a, false, b2, 0, acc2, false, false);
        acc3 = __builtin_amdgcn_wmma_f32_16x16x4_f32(false, a, false, b3, 0, acc3, false, false);
    }

    // C/D layout: VGPR r -> row mBase + r (+8 for lanes >= 16), col = lane&15
    const int mOut = mBase + (lane >> 4) * 8;
    v8f accs[4] = {acc0, acc1, acc2, acc3};
    #pragma unroll
    for (int t = 0; t < 4; ++t) {
        const int nCol = nBase + t * 16 + (lane & 15);
        const float bv = bias[nCol];
        v8f acc = accs[t];
        #pragma unroll
        for (int r = 0; r < 8; ++r) {
            float h = acc[r] + bv;
            h = (h >= 0.0f) ? h : (LEAKY * h);
            Hout[(mOut + r) * HDIM + nCol] = h;
        }
    }
}

// ---------------------------------------------------------------------------
// Kernel 2: cum[b,l,h] = sum_{l'<=l} h1[b,l',h].  One block per b, thread = h.
// ---------------------------------------------------------------------------
__global__ void cumsum_kernel(const float* __restrict__ h1, float* __restrict__ cum)
{
    const int b = blockIdx.x;
    const int h = threadIdx.x;
    const float* src = h1 + b * LDIM * HDIM + h;
    float*       dst = cum + b * LDIM * HDIM + h;
    float run = 0.0f;
    for (int l = 0; l < LDIM; ++l) {
        run += src[l * HDIM];
        dst[l * HDIM] = run;
    }
}

// ---------------------------------------------------------------------------
// Kernel 3: five 512x256x16 projections fused; one wave per 16-row tile.
//   Qj[m,o] = Wf1.h1[m] + 0.5*Wf2.h2[m] + bf[o]
//   Qi[m,o] = 0.5*Wf2.h1[m] - Wf1.h2[m]
//   P5[m,o] = Wf3.cum[m]
// ---------------------------------------------------------------------------
__global__ __launch_bounds__(32) void proj_wmma_kernel(
    const float* __restrict__ h1, const float* __restrict__ h2,
    const float* __restrict__ cum, const float* __restrict__ Wf,
    const float* __restrict__ bf,
    float* __restrict__ Qj, float* __restrict__ Qi, float* __restrict__ P5)
{
    const int lane = threadIdx.x & 31;
    const int mRow = blockIdx.x * 16 + (lane & 15);
    const int kOff = (lane >> 4) * 2;
    const int o    = lane & 15;

    const float* a1p = h1  + mRow * HDIM + kOff;
    const float* a2p = h2  + mRow * HDIM + kOff;
    const float* a5p = cum + mRow * HDIM + kOff;
    const float* w1p = Wf + o * (3 * HDIM) + kOff;          // Wf[:, 0:256]
    const float* w2p = w1p + HDIM;                          // Wf[:, 256:512]
    const float* w3p = w1p + 2 * HDIM;                      // Wf[:, 512:768]

    v8f p1 = {}, p2 = {}, p3 = {}, p4 = {}, p5 = {};
    #pragma unroll 4
    for (int k = 0; k < HDIM; k += 4) {
        v2f a1 = *(const v2f*)(a1p + k);
        v2f a2 = *(const v2f*)(a2p + k);
        v2f a5 = *(const v2f*)(a5p + k);
        v2f b1 = *(const v2f*)(w1p + k);
        v2f b2 = *(const v2f*)(w2p + k);
        v2f b3 = *(const v2f*)(w3p + k);
        p1 = __builtin_amdgcn_wmma_f32_16x16x4_f32(false, a1, false, b1, 0, p1, false, false);
        p2 = __builtin_amdgcn_wmma_f32_16x16x4_f32(false, a2, false, b1, 0, p2, false, false);
        p3 = __builtin_amdgcn_wmma_f32_16x16x4_f32(false, a2, false, b2, 0, p3, false, false);
        p4 = __builtin_amdgcn_wmma_f32_16x16x4_f32(false, a1, false, b2, 0, p4, false, false);
        p5 = __builtin_amdgcn_wmma_f32_16x16x4_f32(false, a5, false, b3, 0, p5, false, false);
    }

    const int mOut = blockIdx.x * 16 + (lane >> 4) * 8;
    const float bfv = bf[o];
    #pragma unroll
    for (int r = 0; r < 8; ++r) {
        const int m = mOut + r;
        Qj[m * ODIM + o] = p1[r] + 0.5f * p3[r] + bfv;
        Qi[m * ODIM + o] = 0.5f * p4[r] - p2[r];
        P5[m * ODIM + o] = p5[r];
    }
}

// ---------------------------------------------------------------------------
// Kernel 4: streaming epilogue, writes out[b,o,i,j].  Block = (i,b), thread = j.
//   8 MB of output -> pure bandwidth; coalesced per-o planes.
// ---------------------------------------------------------------------------
__global__ __launch_bounds__(LDIM) void epilogue_kernel(
    const float* __restrict__ Qj, const float* __restrict__ Qi,
    const float* __restrict__ P5, float* __restrict__ out)
{
    const int i = blockIdx.x;
    const int b = blockIdx.y;
    const int j = threadIdx.x;

    __shared__ float sQi[ODIM];
    __shared__ float sP5p[ODIM];
    if (threadIdx.x < ODIM) {
        const int o = threadIdx.x;
        sQi[o]  = Qi[(b * LDIM + i) * ODIM + o];
        sP5p[o] = (i == 0) ? 0.0f : P5[(b * LDIM + i - 1) * ODIM + o];
    }
    __syncthreads();

    // sizes = (j - i + 1) + eps, computed exactly like the f32 reference
    const float denom = (float)(j - i + 1) + EPSV;
    const float rden  = 1.0f / denom;

    const float* qj = Qj + (b * LDIM + j) * ODIM;
    const float* pj = P5 + (b * LDIM + j) * ODIM;
    float* o0 = out + ((size_t)(b * ODIM) * LDIM + i) * LDIM + j;

    #pragma unroll
    for (int o = 0; o < ODIM; ++o) {
        const float v = qj[o] + sQi[o] + (pj[o] - sP5p[o]) * rden;
        o0[(size_t)o * LDIM * LDIM] = v;
    }
}

// ---------------------------------------------------------------------------
extern "C" void kernel_launch(void* const* d_in, const int* in_sizes, int n_in,
                              void* d_out, int out_size, void* d_ws, size_t ws_size,
                              hipStream_t stream)
{
    const float* x  = (const float*)d_in[0];   // [B,L,D]
    const float* W1 = (const float*)d_in[1];   // [H,D]
    const float* b1 = (const float*)d_in[2];   // [H]
    const float* W2 = (const float*)d_in[3];   // [H,D]
    const float* b2 = (const float*)d_in[4];   // [H]
    const float* Wf = (const float*)d_in[5];   // [O,3H]
    const float* bf = (const float*)d_in[6];   // [O]
    float* out = (float*)d_out;                // [B,O,L,L]

    float* ws  = (float*)d_ws;
    float* h1  = ws;                           // [MDIM, HDIM]
    float* h2  = h1  + (size_t)MDIM * HDIM;
    float* cum = h2  + (size_t)MDIM * HDIM;
    float* Qj  = cum + (size_t)MDIM * HDIM;    // [MDIM, ODIM]
    float* Qi  = Qj  + (size_t)MDIM * ODIM;
    float* P5  = Qi  + (size_t)MDIM * ODIM;

    dim3 g1(MDIM / 16, HDIM / 64);
    mlp_wmma_kernel<<<g1, 32, 0, stream>>>(x, W1, b1, h1);
    mlp_wmma_kernel<<<g1, 32, 0, stream>>>(x, W2, b2, h2);
    cumsum_kernel<<<BDIM, HDIM, 0, stream>>>(h1, cum);
    proj_wmma_kernel<<<MDIM / 16, 32, 0, stream>>>(h1, h2, cum, Wf, bf, Qj, Qi, P5);
    epilogue_kernel<<<dim3(LDIM, BDIM), LDIM, 0, stream>>>(Qj, Qi, P5, out);
}